// GraphCrossModule_75823352644076
// MI455X (gfx1250) — compile-verified
//
#include <hip/hip_runtime.h>
#include <math.h>

// ---------------------------------------------------------------------------
// GraphCrossModule for MI455X (gfx1250, wave32).
//  * Sparse propagation: bandwidth-bound -> float4 gather + atomic f32 scatter.
//  * Dense layers: V_WMMA_F32_16X16X4_F32 (full fp32 matrix path), A staged in
//    LDS (padded stride), one 16x16 tile per wave, 8 waves per block.
//  * top_k: padded bitonic sort, strict (score desc, idx asc) order ==
//    stable top_k. "nonsel" is unused by reference outputs -> skipped.
// Workspace use: ~570 MB inside d_ws (bump allocated).
// ---------------------------------------------------------------------------

typedef float v2f __attribute__((ext_vector_type(2)));
typedef float v8f __attribute__((ext_vector_type(8)));

static __device__ __forceinline__ float warp_sum(float v) {
  for (int off = 16; off > 0; off >>= 1) v += __shfl_down(v, off, 32);
  return v;
}

// ----------------------------- elementwise ---------------------------------
__global__ void k_fill_f32(float* p, float v, int n) {
  int i = blockIdx.x * blockDim.x + threadIdx.x;
  if (i < n) p[i] = v;
}
__global__ void k_fill_i32(int* p, int v, int n) {
  int i = blockIdx.x * blockDim.x + threadIdx.x;
  if (i < n) p[i] = v;
}
__global__ void k_copy_i32(const int* a, int* b, int n) {
  int i = blockIdx.x * blockDim.x + threadIdx.x;
  if (i < n) b[i] = a[i];
}
// y = a + c*b
__global__ void k_add2(float* y, const float* a, float c, const float* b, int n) {
  int i = blockIdx.x * blockDim.x + threadIdx.x;
  if (i < n) y[i] = a[i] + c * b[i];
}
// y = a + c*(b + d)
__global__ void k_add3(float* y, const float* a, float c, const float* b,
                       const float* d, int n) {
  int i = blockIdx.x * blockDim.x + threadIdx.x;
  if (i < n) y[i] = a[i] + c * (b[i] + d[i]);
}

// ----------------------------- graph degree --------------------------------
__global__ void k_deg_lvl1(const int* src, const int* dst, float* deg, int E0) {
  int e = blockIdx.x * blockDim.x + threadIdx.x;
  if (e >= 2 * E0) return;
  int d = (e < E0) ? dst[e] : src[e - E0];
  atomicAdd(&deg[d], 1.0f);
}
__global__ void k_deg_masked(const int* d2, const float* m2, float* deg, int E) {
  int e = blockIdx.x * blockDim.x + threadIdx.x;
  if (e >= E) return;
  float m = m2[e];
  if (m != 0.f) atomicAdd(&deg[d2[e]], m);
}

// ----------------------------- propagation ---------------------------------
// out = x / deg  (self-loop term)
__global__ void k_prop_self(const float* x, const float* deg, float* out,
                            int n, int F) {
  int t = blockIdx.x * blockDim.x + threadIdx.x;
  if (t >= n * F) return;
  out[t] = x[t] / deg[t / F];
}
// out[dst] += rsqrt(deg[s])*rsqrt(deg[d]) * x[src]   (level-1 doubled edges)
__global__ void k_prop_edges_lvl1(const float* x, const int* src, const int* dst,
                                  const float* deg, float* out, int E0, int F4) {
  int t = blockIdx.x * blockDim.x + threadIdx.x;
  if (t >= 2 * E0 * F4) return;
  int e = t / F4, g = t % F4;
  int s = (e < E0) ? src[e] : dst[e - E0];
  int d = (e < E0) ? dst[e] : src[e - E0];
  float w = rsqrtf(deg[s]) * rsqrtf(deg[d]);
  float4 xv = ((const float4*)x)[(size_t)s * F4 + g];
  float* o = out + ((size_t)d * F4 + g) * 4;
  atomicAdd(o + 0, w * xv.x);
  atomicAdd(o + 1, w * xv.y);
  atomicAdd(o + 2, w * xv.z);
  atomicAdd(o + 3, w * xv.w);
}
__global__ void k_prop_edges_masked(const float* x, const int* s2, const int* d2,
                                    const float* m2, const float* deg, float* out,
                                    int E, int F4) {
  int t = blockIdx.x * blockDim.x + threadIdx.x;
  if (t >= E * F4) return;
  int e = t / F4, g = t % F4;
  float m = m2[e];
  if (m == 0.f) return;
  int s = s2[e], d = d2[e];
  float w = m * rsqrtf(deg[s]) * rsqrtf(deg[d]);
  float4 xv = ((const float4*)x)[(size_t)s * F4 + g];
  float* o = out + ((size_t)d * F4 + g) * 4;
  atomicAdd(o + 0, w * xv.x);
  atomicAdd(o + 1, w * xv.y);
  atomicAdd(o + 2, w * xv.z);
  atomicAdd(o + 3, w * xv.w);
}

// ----------------------------- pooled-graph build --------------------------
__global__ void k_build_edges_lvl1(const int* src, const int* dst, const int* inv,
                                   int* sO, int* dO, float* mO, int E0) {
  int e = blockIdx.x * blockDim.x + threadIdx.x;
  if (e >= 2 * E0) return;
  int s0 = (e < E0) ? src[e] : dst[e - E0];
  int d0 = (e < E0) ? dst[e] : src[e - E0];
  int is_ = inv[s0], id_ = inv[d0];
  sO[e] = is_ < 0 ? 0 : is_;
  dO[e] = id_ < 0 ? 0 : id_;
  mO[e] = (is_ >= 0 && id_ >= 0) ? 1.0f : 0.0f;
}
__global__ void k_build_edges_gen(const int* sI, const int* dI, const float* mI,
                                  const int* inv, int* sO, int* dO, float* mO,
                                  int E) {
  int e = blockIdx.x * blockDim.x + threadIdx.x;
  if (e >= E) return;
  int is_ = inv[sI[e]], id_ = inv[dI[e]];
  sO[e] = is_ < 0 ? 0 : is_;
  dO[e] = id_ < 0 ? 0 : id_;
  mO[e] = (is_ >= 0 && id_ >= 0) ? mI[e] : 0.0f;
}

// ----------------------------- WMMA GEMM -----------------------------------
// C(Mx128) = act( [A0|A1|A2](M x nparts*F) @ W(K x 128) + bias )
// act: 0 none, 1 relu, 2 prelu(prelu_a[0]).  In-place (C==A0) is safe:
// each block stages its 16 A-rows in LDS before writing those same rows.
__global__ __launch_bounds__(256)
void k_gemm_wmma(const float* A0, const float* A1, const float* A2, int F,
                 int nparts, const float* Wm, const float* bias, float* C,
                 int M, int act, const float* prelu_a) {
  __shared__ float lA[16 * (768 + 4)];
  const int K = F * nparts;
  const int KP = K + 4;  // pad: stride mod 64 == 4 -> conflict-free reads
  const int rowBase = blockIdx.x * 16;
  const int tid = threadIdx.x;

  for (int idx = tid; idx < 16 * K; idx += 256) {
    int r = idx / K, k = idx % K;
    int row = rowBase + r;
    float v = 0.f;
    if (row < M) {
      int part = k / F, col = k - part * F;
      const float* A = (part == 0) ? A0 : ((part == 1) ? A1 : A2);
      v = A[(size_t)row * F + col];
    }
    lA[r * KP + k] = v;
  }
  __syncthreads();

  const int wave = tid >> 5, lane = tid & 31;
  const int half = lane >> 4, li = lane & 15;
  const int colBase = wave * 16;

  v8f acc;
  for (int r = 0; r < 8; ++r) acc[r] = 0.f;

  for (int k0 = 0; k0 < K; k0 += 4) {
    v2f a, b;
    a.x = lA[li * KP + k0 + 2 * half];
    a.y = lA[li * KP + k0 + 2 * half + 1];
    b.x = Wm[(size_t)(k0 + 2 * half) * 128 + colBase + li];
    b.y = Wm[(size_t)(k0 + 2 * half + 1) * 128 + colBase + li];
    acc = __builtin_amdgcn_wmma_f32_16x16x4_f32(false, a, false, b, (short)0,
                                                acc, false, false);
  }

  float pa = (act == 2) ? prelu_a[0] : 0.f;
  float bv = bias[colBase + li];
  for (int r = 0; r < 8; ++r) {
    int row = rowBase + r + 8 * half;
    if (row < M) {
      float v = acc[r] + bv;
      if (act == 1) v = v > 0.f ? v : 0.f;
      else if (act == 2) v = v >= 0.f ? v : pa * v;
      C[(size_t)row * 128 + colBase + li] = v;
    }
  }
}

// ----------------------------- row reductions ------------------------------
__global__ void k_rowdot(const float* a, const float* b, float* out, int M) {
  int warp = (blockIdx.x * blockDim.x + threadIdx.x) >> 5;
  int lane = threadIdx.x & 31;
  if (warp >= M) return;
  const float* ar = a + (size_t)warp * 128;
  const float* br = b + (size_t)warp * 128;
  float s = 0.f;
  for (int f = lane; f < 128; f += 32) s += ar[f] * br[f];
  s = warp_sum(s);
  if (lane == 0) out[warp] = s;
}
__global__ void k_rowabsdiff(const float* a, const float* b, float* out, int M) {
  int warp = (blockIdx.x * blockDim.x + threadIdx.x) >> 5;
  int lane = threadIdx.x & 31;
  if (warp >= M) return;
  const float* ar = a + (size_t)warp * 128;
  const float* br = b + (size_t)warp * 128;
  float s = 0.f;
  for (int f = lane; f < 128; f += 32) s += fabsf(ar[f] - br[f]);
  s = warp_sum(s);
  if (lane == 0) out[warp] = s;
}
__global__ void k_absmax(const float* x, int n, float* maxv) {
  int i = blockIdx.x * blockDim.x + threadIdx.x;
  if (i >= n) return;
  atomicMax((int*)maxv, __float_as_int(x[i]));  // x >= 0
}
__global__ void k_scale_scores(float* sc, const float* info, const float* maxv,
                               int n) {
  int i = blockIdx.x * blockDim.x + threadIdx.x;
  if (i < n) sc[i] = info[i] / (maxv[0] + 1e-8f);
}

// ----------------------------- sort (top-k) --------------------------------
__global__ void k_sort_init(float* key, int* idx, const float* sc, int n,
                            int pad) {
  int i = blockIdx.x * blockDim.x + threadIdx.x;
  if (i >= pad) return;
  key[i] = (i < n) ? sc[i] : -INFINITY;
  idx[i] = i;
}
__global__ void k_bitonic(float* key, int* idx, int j, int k) {
  int i = blockIdx.x * blockDim.x + threadIdx.x;
  int ixj = i ^ j;
  if (ixj <= i) return;
  float ki = key[i], kj = key[ixj];
  int ii = idx[i], ij = idx[ixj];
  bool before = (ki > kj) || (ki == kj && ii < ij);  // i precedes ixj?
  bool up = ((i & k) == 0);
  if (up ? !before : before) {
    key[i] = kj; key[ixj] = ki;
    idx[i] = ij; idx[ixj] = ii;
  }
}
__global__ void k_scatter_invmap(const int* sel, int* inv, int K) {
  int j = blockIdx.x * blockDim.x + threadIdx.x;
  if (j < K) inv[sel[j]] = j;
}

// ----------------------------- gather / scatter ----------------------------
__global__ void k_gather_rows(const float* x, const int* idx, float* out,
                              int Mout) {
  int t = blockIdx.x * blockDim.x + threadIdx.x;
  if (t >= Mout * 128) return;
  int r = t >> 7, f = t & 127;
  out[t] = x[(size_t)idx[r] * 128 + f];
}
__global__ void k_gather_scale(const float* x, const int* sel, const float* sc,
                               float* out, int K) {
  int t = blockIdx.x * blockDim.x + threadIdx.x;
  if (t >= K * 128) return;
  int r = t >> 7, f = t & 127;
  int n = sel[r];
  out[t] = x[(size_t)n * 128 + f] * sc[n];
}
__global__ void k_scale_rows(const float* x, const float* sc, float* out, int n) {
  int t = blockIdx.x * blockDim.x + threadIdx.x;
  if (t >= n * 128) return;
  out[t] = x[t] * sc[t >> 7];
}
__global__ void k_scatter_rows(const float* xc, const int* sel, float* out,
                               int K) {
  int t = blockIdx.x * blockDim.x + threadIdx.x;
  if (t >= K * 128) return;
  int r = t >> 7, f = t & 127;
  out[(size_t)sel[r] * 128 + f] = xc[t];
}
__global__ void k_concat_cols(const float* a, const float* b, float* out, int n) {
  int t = blockIdx.x * blockDim.x + threadIdx.x;
  if (t >= n * 256) return;
  int r = t >> 8, c = t & 255;
  out[t] = (c < 128) ? a[(size_t)r * 128 + c] : b[(size_t)r * 128 + c - 128];
}

// ---------------------------------------------------------------------------
extern "C" void kernel_launch(void* const* d_in, const int* in_sizes, int n_in,
                              void* d_out, int out_size, void* d_ws,
                              size_t ws_size, hipStream_t stream) {
  (void)in_sizes; (void)n_in; (void)out_size; (void)ws_size;
  constexpr int N1 = 40000, E0 = 640000, E2 = 2 * E0;
  constexpr int K1 = 32000, K2 = 25600;           // ceil(0.8*N1), ceil(0.8*K1)
  constexpr int PAD1 = 65536, PAD2 = 32768;
  constexpr size_t NF = (size_t)N1 * 128, N2F = (size_t)N1 * 256;
  constexpr size_t K1F = (size_t)K1 * 128, K2F = (size_t)K2 * 128;

  // ---- inputs (flattened setup_inputs dict order) ----
  const float* feat  = (const float*)d_in[0];
  const float* sg1W  = (const float*)d_in[1];  const float* sg1b  = (const float*)d_in[2];
  const float* sg2W  = (const float*)d_in[3];  const float* sg2b  = (const float*)d_in[4];
  const float* endW  = (const float*)d_in[5];  const float* endb  = (const float*)d_in[6];
  const float* s1l1W = (const float*)d_in[7];  const float* s1l1b = (const float*)d_in[8];
  const float* s1l2W = (const float*)d_in[9];  const float* s1l2b = (const float*)d_in[10];
  const float* s2l1W = (const float*)d_in[11]; const float* s2l1b = (const float*)d_in[12];
  const float* s2l2W = (const float*)d_in[13]; const float* s2l2b = (const float*)d_in[14];
  const float* s3l1W = (const float*)d_in[15]; const float* s3l1b = (const float*)d_in[16];
  const float* s3l2W = (const float*)d_in[17]; const float* s3l2b = (const float*)d_in[18];
  const float* is1W  = (const float*)d_in[19]; const float* is1b  = (const float*)d_in[20];
  const float* is1a  = (const float*)d_in[21];
  const float* is2W  = (const float*)d_in[22]; const float* is2b  = (const float*)d_in[23];
  const float* is2a  = (const float*)d_in[24];
  const float* up21W = (const float*)d_in[25]; const float* up21b = (const float*)d_in[26];
  const float* up32W = (const float*)d_in[27]; const float* up32b = (const float*)d_in[28];
  const float* p12W  = (const float*)d_in[29]; const float* p12b  = (const float*)d_in[30];
  const float* p23W  = (const float*)d_in[31]; const float* p23b  = (const float*)d_in[32];
  const float* u21W  = (const float*)d_in[33]; const float* u21b  = (const float*)d_in[34];
  const float* u32W  = (const float*)d_in[35]; const float* u32b  = (const float*)d_in[36];
  const float* cf1W  = (const float*)d_in[37]; const float* cf1b  = (const float*)d_in[38];
  const int* src   = (const int*)d_in[39];
  const int* dst   = (const int*)d_in[40];
  const int* perm1 = (const int*)d_in[41];
  const int* perm2 = (const int*)d_in[42];

  float* out    = (float*)d_out;
  float* outAgg = out;                       // N1*128
  float* outL1  = out + NF;                  // 2*N1
  float* outL2  = outL1 + 2 * (size_t)N1;    // 2*K1

  // ---- workspace bump allocator ----
  size_t off = 0;
  auto alloc_f = [&](size_t n) -> float* {
    off = (off + 255) & ~(size_t)255;
    float* p = (float*)((char*)d_ws + off);
    off += n * sizeof(float);
    return p;
  };
  auto alloc_i = [&](size_t n) -> int* {
    off = (off + 255) & ~(size_t)255;
    int* p = (int*)((char*)d_ws + off);
    off += n * sizeof(int);
    return p;
  };
  float *F1 = alloc_f(NF), *DOWN1 = alloc_f(NF), *F1b = alloc_f(NF);
  float *F1c = alloc_f(NF), *F1d = alloc_f(NF), *F2OUT = alloc_f(NF);
  float *TA = alloc_f(N2F), *TB = alloc_f(N2F), *TG = alloc_f(N2F);
  float *TC = alloc_f(NF), *TD = alloc_f(NF);
  float *F2 = alloc_f(K1F), *F2b = alloc_f(K1F), *F2c = alloc_f(K1F);
  float *F2d = alloc_f(K1F), *F2e = alloc_f(K1F), *DOWN2 = alloc_f(K1F);
  float *F12 = alloc_f(K1F), *F32b = alloc_f(K1F), *F3OUT = alloc_f(K1F);
  float *TK = alloc_f(K1F), *TK2 = alloc_f(K1F);
  float *F3 = alloc_f(K2F), *F3b = alloc_f(K2F), *F3c = alloc_f(K2F);
  float *F3d = alloc_f(K2F), *F23 = alloc_f(K2F);
  float *TL = alloc_f(K2F), *TL2 = alloc_f(K2F);
  float *deg1 = alloc_f(N1), *deg2 = alloc_f(K1), *deg3 = alloc_f(K2);
  float *info1 = alloc_f(N1), *sc1 = alloc_f(N1);
  float *info2 = alloc_f(K1), *sc2 = alloc_f(K1);
  float *maxv = alloc_f(64);
  float *EMSK2 = alloc_f(E2), *EMSK3 = alloc_f(E2);  // edge masks (avoid HIP MASK2 macro)
  float *skey = alloc_f(PAD1);
  int *SRC2 = alloc_i(E2), *DST2 = alloc_i(E2);
  int *SRC3 = alloc_i(E2), *DST3 = alloc_i(E2);
  int *sidx = alloc_i(PAD1);
  int *sel1 = alloc_i(K1), *sel2 = alloc_i(K2);
  int *inv1 = alloc_i(N1), *inv2 = alloc_i(K1);

  auto cdiv = [](long a, long b) { return (int)((a + b - 1) / b); };
  const int B = 256;

  // ---- helpers ----
  auto prop1 = [&](const float* x, float* o, int F) {
    int F4 = F / 4;
    k_prop_self<<<cdiv((long)N1 * F, B), B, 0, stream>>>(x, deg1, o, N1, F);
    k_prop_edges_lvl1<<<cdiv((long)E2 * F4, B), B, 0, stream>>>(x, src, dst,
                                                                deg1, o, E0, F4);
  };
  auto prop2 = [&](const float* x, float* o) {
    k_prop_self<<<cdiv((long)K1 * 128, B), B, 0, stream>>>(x, deg2, o, K1, 128);
    k_prop_edges_masked<<<cdiv((long)E2 * 32, B), B, 0, stream>>>(
        x, SRC2, DST2, EMSK2, deg2, o, E2, 32);
  };
  auto prop3 = [&](const float* x, float* o) {
    k_prop_self<<<cdiv((long)K2 * 128, B), B, 0, stream>>>(x, deg3, o, K2, 128);
    k_prop_edges_masked<<<cdiv((long)E2 * 32, B), B, 0, stream>>>(
        x, SRC3, DST3, EMSK3, deg3, o, E2, 32);
  };
  auto gemm = [&](const float* A0, const float* A1, const float* A2, int F,
                  int np, const float* Wm, const float* bb, float* Cc, int M,
                  int act, const float* pa) {
    k_gemm_wmma<<<cdiv(M, 16), 256, 0, stream>>>(A0, A1, A2, F, np, Wm, bb, Cc,
                                                 M, act, pa);
  };
  auto sortsel = [&](const float* sc, int n, int pad, int* sel, int Ksel) {
    k_sort_init<<<cdiv(pad, B), B, 0, stream>>>(skey, sidx, sc, n, pad);
    for (int k = 2; k <= pad; k <<= 1)
      for (int j = k >> 1; j > 0; j >>= 1)
        k_bitonic<<<cdiv(pad, B), B, 0, stream>>>(skey, sidx, j, k);
    k_copy_i32<<<cdiv(Ksel, B), B, 0, stream>>>(sidx, sel, Ksel);
  };

  // ================= level-1 graph =================
  k_fill_f32<<<cdiv(N1, B), B, 0, stream>>>(deg1, 1.0f, N1);
  k_deg_lvl1<<<cdiv(E2, B), B, 0, stream>>>(src, dst, deg1, E0);

  // f1 = tag(prop1, feat, sg1)
  prop1(feat, TC, 128);
  prop1(TC, TD, 128);
  gemm(feat, TC, TD, 128, 3, sg1W, sg1b, F1, N1, 0, nullptr);

  // index_select level 1
  prop1(F1, TC, 128);                                        // s
  k_rowabsdiff<<<cdiv((long)N1 * 32, B), B, 0, stream>>>(F1, TC, info1, N1);
  k_fill_f32<<<1, 64, 0, stream>>>(maxv, 0.0f, 64);
  k_absmax<<<cdiv(N1, B), B, 0, stream>>>(info1, N1, maxv);
  k_scale_scores<<<cdiv(N1, B), B, 0, stream>>>(sc1, info1, maxv, N1);
  gemm(F1, nullptr, nullptr, 128, 1, is1W, is1b, TD, N1, 2, is1a);   // pos
  k_rowdot<<<cdiv((long)N1 * 32, B), B, 0, stream>>>(TD, TC, outL1, N1);
  k_gather_rows<<<cdiv(NF, B), B, 0, stream>>>(F1, perm1, TD, N1);
  gemm(TD, nullptr, nullptr, 128, 1, is1W, is1b, TD, N1, 2, is1a);   // neg
  k_rowdot<<<cdiv((long)N1 * 32, B), B, 0, stream>>>(TD, TC, outL1 + N1, N1);
  sortsel(sc1, N1, PAD1, sel1, K1);
  k_scale_rows<<<cdiv(NF, B), B, 0, stream>>>(F1, sc1, DOWN1, N1);   // down1

  // pool -> level 2
  k_gather_scale<<<cdiv(K1F, B), B, 0, stream>>>(F1, sel1, sc1, F2, K1);
  k_fill_i32<<<cdiv(N1, B), B, 0, stream>>>(inv1, -1, N1);
  k_scatter_invmap<<<cdiv(K1, B), B, 0, stream>>>(sel1, inv1, K1);
  k_build_edges_lvl1<<<cdiv(E2, B), B, 0, stream>>>(src, dst, inv1, SRC2, DST2,
                                                    EMSK2, E0);
  k_fill_f32<<<cdiv(K1, B), B, 0, stream>>>(deg2, 1.0f, K1);
  k_deg_masked<<<cdiv(E2, B), B, 0, stream>>>(DST2, EMSK2, deg2, E2);

  // f2 = tag(prop2, F2, sg2)
  prop2(F2, TK);
  prop2(TK, TK2);
  gemm(F2, TK, TK2, 128, 3, sg2W, sg2b, F2b, K1, 0, nullptr);

  // index_select level 2
  prop2(F2b, TK);
  k_rowabsdiff<<<cdiv((long)K1 * 32, B), B, 0, stream>>>(F2b, TK, info2, K1);
  k_fill_f32<<<1, 64, 0, stream>>>(maxv, 0.0f, 64);
  k_absmax<<<cdiv(K1, B), B, 0, stream>>>(info2, K1, maxv);
  k_scale_scores<<<cdiv(K1, B), B, 0, stream>>>(sc2, info2, maxv, K1);
  gemm(F2b, nullptr, nullptr, 128, 1, is2W, is2b, TK2, K1, 2, is2a);
  k_rowdot<<<cdiv((long)K1 * 32, B), B, 0, stream>>>(TK2, TK, outL2, K1);
  k_gather_rows<<<cdiv(K1F, B), B, 0, stream>>>(F2b, perm2, TK2, K1);
  gemm(TK2, nullptr, nullptr, 128, 1, is2W, is2b, TK2, K1, 2, is2a);
  k_rowdot<<<cdiv((long)K1 * 32, B), B, 0, stream>>>(TK2, TK, outL2 + K1, K1);
  sortsel(sc2, K1, PAD2, sel2, K2);
  k_scale_rows<<<cdiv(K1F, B), B, 0, stream>>>(F2b, sc2, DOWN2, K1);  // down2

  // pool -> level 3
  k_gather_scale<<<cdiv(K2F, B), B, 0, stream>>>(F2b, sel2, sc2, F3, K2);
  k_fill_i32<<<cdiv(K1, B), B, 0, stream>>>(inv2, -1, K1);
  k_scatter_invmap<<<cdiv(K2, B), B, 0, stream>>>(sel2, inv2, K2);
  k_build_edges_gen<<<cdiv(E2, B), B, 0, stream>>>(SRC2, DST2, EMSK2, inv2,
                                                   SRC3, DST3, EMSK3, E2);
  k_fill_f32<<<cdiv(K2, B), B, 0, stream>>>(deg3, 1.0f, K2);
  k_deg_masked<<<cdiv(E2, B), B, 0, stream>>>(DST3, EMSK3, deg3, E2);

  // ---- stage-1 tags with relu ----
  prop1(F1, TC, 128);  prop1(TC, TD, 128);
  gemm(F1, TC, TD, 128, 3, s1l1W, s1l1b, F1b, N1, 1, nullptr);
  prop2(F2b, TK);      prop2(TK, TK2);
  gemm(F2b, TK, TK2, 128, 3, s2l1W, s2l1b, F2c, K1, 1, nullptr);
  prop3(F3, TL);       prop3(TL, TL2);
  gemm(F3, TL, TL2, 128, 3, s3l1W, s3l1b, F3b, K2, 1, nullptr);

  // ---- cross terms ----
  // f12 = pool(prop1, f1; p12)
  prop1(F1b, TC, 128);
  gemm(TC, nullptr, nullptr, 128, 1, p12W, p12b, TD, N1, 0, nullptr);
  k_gather_scale<<<cdiv(K1F, B), B, 0, stream>>>(TD, sel1, sc1, F12, K1);
  // f21 = unpool(prop1, f2; u21)
  k_fill_f32<<<cdiv(NF, B), B, 0, stream>>>(TC, 0.0f, (int)NF);
  k_scatter_rows<<<cdiv(K1F, B), B, 0, stream>>>(F2c, sel1, TC, K1);
  prop1(TC, TD, 128);
  gemm(TD, nullptr, nullptr, 128, 1, u21W, u21b, TC, N1, 0, nullptr);
  k_add2<<<cdiv(NF, B), B, 0, stream>>>(F1b, F1b, 1.0f, TC, (int)NF);
  // f23 = pool(prop2, f2; p23)
  prop2(F2c, TK);
  gemm(TK, nullptr, nullptr, 128, 1, p23W, p23b, TK2, K1, 0, nullptr);
  k_gather_scale<<<cdiv(K2F, B), B, 0, stream>>>(TK2, sel2, sc2, F23, K2);
  // f32 = unpool(prop2, f3; u32)
  k_fill_f32<<<cdiv(K1F, B), B, 0, stream>>>(TK, 0.0f, (int)K1F);
  k_scatter_rows<<<cdiv(K2F, B), B, 0, stream>>>(F3b, sel2, TK, K2);
  prop2(TK, TK2);
  gemm(TK2, nullptr, nullptr, 128, 1, u32W, u32b, F32b, K1, 0, nullptr);
  // updates
  k_add3<<<cdiv(K1F, B), B, 0, stream>>>(F2c, F2c, 0.5f, F12, F32b, (int)K1F);
  k_add2<<<cdiv(K2F, B), B, 0, stream>>>(F3b, F3b, 1.0f, F23, (int)K2F);

  // ---- cf1 on the concatenation (per-part GEMM) ----
  gemm(F1b, nullptr, nullptr, 128, 1, cf1W, cf1b, F1c, N1, 0, nullptr);
  gemm(F2c, nullptr, nullptr, 128, 1, cf1W, cf1b, F2d, K1, 0, nullptr);
  gemm(F3b, nullptr, nullptr, 128, 1, cf1W, cf1b, F3c, K2, 0, nullptr);

  // ---- stage-2 tags with relu ----
  prop1(F1c, TC, 128); prop1(TC, TD, 128);
  gemm(F1c, TC, TD, 128, 3, s1l2W, s1l2b, F1d, N1, 1, nullptr);
  prop2(F2d, TK);      prop2(TK, TK2);
  gemm(F2d, TK, TK2, 128, 3, s2l2W, s2l2b, F2e, K1, 1, nullptr);
  prop3(F3c, TL);      prop3(TL, TL2);
  gemm(F3c, TL, TL2, 128, 3, s3l2W, s3l2b, F3d, K2, 1, nullptr);

  // ---- decoder ----
  // f3_out = unpool(prop2, f3; up32) + down2
  k_fill_f32<<<cdiv(K1F, B), B, 0, stream>>>(TK, 0.0f, (int)K1F);
  k_scatter_rows<<<cdiv(K2F, B), B, 0, stream>>>(F3d, sel2, TK, K2);
  prop2(TK, TK2);
  gemm(TK2, nullptr, nullptr, 128, 1, up32W, up32b, F3OUT, K1, 0, nullptr);
  k_add2<<<cdiv(K1F, B), B, 0, stream>>>(F3OUT, F3OUT, 1.0f, DOWN2, (int)K1F);
  // f2_out = unpool(prop1, f2 + f3_out; up21)
  k_add2<<<cdiv(K1F, B), B, 0, stream>>>(TK, F2e, 1.0f, F3OUT, (int)K1F);
  k_fill_f32<<<cdiv(NF, B), B, 0, stream>>>(TC, 0.0f, (int)NF);
  k_scatter_rows<<<cdiv(K1F, B), B, 0, stream>>>(TK, sel1, TC, K1);
  prop1(TC, TD, 128);
  gemm(TD, nullptr, nullptr, 128, 1, up21W, up21b, F2OUT, N1, 0, nullptr);
  // agg = f1 + f2_out + down1
  k_add3<<<cdiv(NF, B), B, 0, stream>>>(F1d, F1d, 1.0f, F2OUT, DOWN1, (int)NF);

  // ---- final: agg256 = [agg | feat_origin]; out = tag(prop1, agg256, end) ----
  k_concat_cols<<<cdiv(N2F, B), B, 0, stream>>>(F1d, F1, TA, N1);
  prop1(TA, TB, 256);
  prop1(TB, TG, 256);
  gemm(TA, TB, TG, 256, 3, endW, endb, outAgg, N1, 0, nullptr);
}